// Lasso_Compressive_81913616269645
// MI455X (gfx1250) — compile-verified
//
#include <hip/hip_runtime.h>
#include <math.h>

// ---------------------------------------------------------------------------
// Lasso / FISTA compressive sensing, MI455X (gfx1250) version.
//
// Bandwidth analysis: coding_mat = 8 x 256 x 65536 f32 = 536 MB. Everything is
// HBM-bound (23.3 TB/s => ~23us per full pass over A). Reference needs ~122
// passes. We cut the 20 power-iteration passes to 1 by power-iterating on the
// 256x256 Gram matrix G = A A^T (computed with V_WMMA_F32_16X16X4_F32, one
// streaming pass over A), which is mathematically identical:
//   p_k = A v_k ;  p_{k+1} = G p_k / (sqrt(p_k^T G p_k) + eps) ; L = |p_10|^2/M
// The 50 FISTA steps are irreducibly 2 passes each (soft-threshold is
// elementwise in D-space), implemented as coalesced streaming kernels.
// ---------------------------------------------------------------------------

#define N_ITERS 50
#define N_POWER 10
#define BATCH   8
#define M_DIM   256
#define D_DIM   65536
#define ALPHA_F 0.5f

typedef float v2f __attribute__((ext_vector_type(2)));
typedef float v8f __attribute__((ext_vector_type(8)));

// ---------------------------------------------------------------------------
// Kernel 1: one pass over A computing BOTH y[b,m] = <A_row, x_b> and
// p0[b,m] = rowsum(A_row)/sqrt(D)   (since v0 = ones/sqrt(D), p0 = A v0).
// One 256-thread block per (b,m) row; float4 streaming loads.
// ---------------------------------------------------------------------------
__global__ __launch_bounds__(256) void prep_rows_kernel(
    const float* __restrict__ A, const float* __restrict__ x,
    float* __restrict__ y, float* __restrict__ p) {
  const int blk = blockIdx.x;            // b*256 + m
  const int b   = blk >> 8;
  const int tid = threadIdx.x;
  const float4* Arow = (const float4*)(A + (size_t)blk * D_DIM);
  const float4* xb   = (const float4*)(x + (size_t)b * D_DIM);
  float accd = 0.f, accs = 0.f;
#pragma unroll 4
  for (int it = 0; it < D_DIM / 4 / 256; ++it) {
    const int i = it * 256 + tid;
    const float4 a  = Arow[i];
    const float4 xv = xb[i];
    accd += a.x * xv.x + a.y * xv.y + a.z * xv.z + a.w * xv.w;
    accs += a.x + a.y + a.z + a.w;
  }
  __shared__ float red[512];
  red[tid] = accd; red[tid + 256] = accs;
  __syncthreads();
  for (int s = 128; s > 0; s >>= 1) {
    if (tid < s) { red[tid] += red[tid + s]; red[tid + 256] += red[tid + 256 + s]; }
    __syncthreads();
  }
  if (tid == 0) {
    y[blk] = red[0];
    p[blk] = red[256] * (1.0f / 256.0f);   // 1/sqrt(65536)
  }
}

// ---------------------------------------------------------------------------
// Kernel 2: Gram matrix G[b] = A_b * A_b^T  (256x256 per batch) via
// V_WMMA_F32_16X16X4_F32. One wave per 16x16 output tile (2048 tiles total,
// 8 waves per block). Panels of A (16 rows x 32 k-cols) are staged in
// per-wave LDS with stride 33 (bank-conflict padding), then 8 WMMAs per
// 32-wide k-chunk. Fragment layout per ISA 7.12.2 (f32 A 16x4):
//   lanes 0-15 : M=lane,    VGPR0=K0, VGPR1=K1
//   lanes 16-31: M=lane-16, VGPR0=K2, VGPR1=K3
// ---------------------------------------------------------------------------
__global__ __launch_bounds__(256) void gram_kernel(
    const float* __restrict__ A, float* __restrict__ G) {
  __shared__ float lds[8][2][16 * 33];
  const int tid  = threadIdx.x;
  const int wave = tid >> 5;
  const int lane = tid & 31;
  const int tile = blockIdx.x * 8 + wave;   // 0..2047
  const int b  = tile >> 8;
  const int t  = tile & 255;
  const int i0 = (t >> 4) << 4;
  const int j0 = (t & 15) << 4;
  const float* Ab = A + (size_t)b * M_DIM * D_DIM;
  float* Pi = &lds[wave][0][0];
  float* Pj = &lds[wave][1][0];

  const int mrow  = lane & 15;
  const int khalf = (lane >> 4) << 1;       // 0 for lanes 0-15, 2 for 16-31
  v8f c = {};

  for (int d0 = 0; d0 < D_DIM; d0 += 32) {
    // Stage two 16x32 panels (coalesced 128B row segments).
#pragma unroll 4
    for (int r = 0; r < 16; ++r) {
      Pi[r * 33 + lane] = Ab[(size_t)(i0 + r) * D_DIM + d0 + lane];
      Pj[r * 33 + lane] = Ab[(size_t)(j0 + r) * D_DIM + d0 + lane];
    }
    __syncthreads();
#pragma unroll
    for (int kk = 0; kk < 32; kk += 4) {
      v2f a, bb;
      a.x  = Pi[mrow * 33 + kk + khalf];
      a.y  = Pi[mrow * 33 + kk + khalf + 1];
      bb.x = Pj[mrow * 33 + kk + khalf];
      bb.y = Pj[mrow * 33 + kk + khalf + 1];
#if __has_builtin(__builtin_amdgcn_wmma_f32_16x16x4_f32)
      c = __builtin_amdgcn_wmma_f32_16x16x4_f32(false, a, false, bb,
                                                (short)0, c, false, false);
#else
      asm volatile("v_wmma_f32_16x16x4_f32 %0, %1, %2, %0"
                   : "+v"(c) : "v"(a), "v"(bb));
#endif
    }
    __syncthreads();
  }

  // C/D layout: VGPR r -> M = i0 + r + (lane>=16 ? 8 : 0), N = j0 + (lane&15)
  const int col   = j0 + (lane & 15);
  const int rbase = i0 + ((lane >> 4) << 3);
  float* Gb = G + ((size_t)b << 16);
#pragma unroll
  for (int rr = 0; rr < 8; ++rr)
    Gb[(size_t)(rbase + rr) * M_DIM + col] = c[rr];
}

// ---------------------------------------------------------------------------
// Kernel 3: power iteration entirely in M-space on G (256x256, L2-hot).
// p_{k+1} = G p_k / (sqrt(p^T G p) + 1e-12);  L = |p|^2 / M after 10 iters.
// One block per batch.
// ---------------------------------------------------------------------------
__global__ __launch_bounds__(256) void power_kernel(
    const float* __restrict__ G, const float* __restrict__ p_in,
    float* __restrict__ L) {
  const int b   = blockIdx.x;
  const int tid = threadIdx.x;
  __shared__ float ps[256];
  __shared__ float red[256];
  const float* Gb = G + ((size_t)b << 16);
  ps[tid] = p_in[b * 256 + tid];
  __syncthreads();
  for (int it = 0; it < N_POWER; ++it) {
    float q = 0.f;
    const float* grow = Gb + (size_t)tid * 256;
#pragma unroll 8
    for (int j = 0; j < 256; ++j) q += grow[j] * ps[j];
    red[tid] = q * ps[tid];                 // p^T (G p)
    __syncthreads();
    for (int s = 128; s > 0; s >>= 1) {
      if (tid < s) red[tid] += red[tid + s];
      __syncthreads();
    }
    const float denom = sqrtf(red[0]) + 1e-12f;
    __syncthreads();
    ps[tid] = q / denom;
    __syncthreads();
  }
  red[tid] = ps[tid] * ps[tid];
  __syncthreads();
  for (int s = 128; s > 0; s >>= 1) {
    if (tid < s) red[tid] += red[tid + s];
    __syncthreads();
  }
  if (tid == 0) L[b] = red[0] * (1.0f / M_DIM);
}

// ---------------------------------------------------------------------------
// Kernel 4: zero-init w (lives in d_out) and z.
// ---------------------------------------------------------------------------
__global__ __launch_bounds__(256) void init_wz_kernel(float* __restrict__ w,
                                                      float* __restrict__ z) {
  const size_t i = (size_t)blockIdx.x * 256 + threadIdx.x;
  w[i] = 0.f; z[i] = 0.f;
}

// ---------------------------------------------------------------------------
// Kernel 5: r[b,m] = <A_row, z_b> - y[b,m].  One block per (b,m) row.
// ---------------------------------------------------------------------------
__global__ __launch_bounds__(256) void residual_kernel(
    const float* __restrict__ A, const float* __restrict__ z,
    const float* __restrict__ y, float* __restrict__ r) {
  const int blk = blockIdx.x;
  const int b   = blk >> 8;
  const int tid = threadIdx.x;
  const float4* Arow = (const float4*)(A + (size_t)blk * D_DIM);
  const float4* zb   = (const float4*)(z + (size_t)b * D_DIM);
  float acc = 0.f;
#pragma unroll 4
  for (int it = 0; it < D_DIM / 4 / 256; ++it) {
    const int i = it * 256 + tid;
    const float4 a  = Arow[i];
    const float4 zv = zb[i];
    acc += a.x * zv.x + a.y * zv.y + a.z * zv.z + a.w * zv.w;
  }
  __shared__ float red[256];
  red[tid] = acc;
  __syncthreads();
  for (int s = 128; s > 0; s >>= 1) {
    if (tid < s) red[tid] += red[tid + s];
    __syncthreads();
  }
  if (tid == 0) r[blk] = red[0] - y[blk];
}

// ---------------------------------------------------------------------------
// Kernel 6: fused  grad = A^T r / M ; w_new = soft(z - grad/L, alpha/L) ;
// z = w_new + coef*(w_new - w) ; w = w_new.
// 64 blocks per batch; each thread owns 4 contiguous d's (float4 streaming,
// fully coalesced 4KB row segments per block per m).
// ---------------------------------------------------------------------------
__global__ __launch_bounds__(256) void update_kernel(
    const float* __restrict__ A, const float* __restrict__ rvec,
    const float* __restrict__ L, float* __restrict__ w,
    float* __restrict__ z, float coef) {
  const int b     = blockIdx.x >> 6;
  const int chunk = blockIdx.x & 63;
  const int tid   = threadIdx.x;
  __shared__ float rs[256];
  rs[tid] = rvec[b * 256 + tid];
  __syncthreads();

  const int d0 = chunk * 1024 + tid * 4;
  const float* Ab = A + (size_t)b * M_DIM * D_DIM + d0;
  float4 acc = make_float4(0.f, 0.f, 0.f, 0.f);
#pragma unroll 4
  for (int m = 0; m < M_DIM; ++m) {
    const float4 av = *(const float4*)(Ab + (size_t)m * D_DIM);
    const float rm = rs[m];
    acc.x += av.x * rm; acc.y += av.y * rm;
    acc.z += av.z * rm; acc.w += av.w * rm;
  }
  const float invL   = 1.0f / L[b];
  const float thr    = ALPHA_F * invL;
  const float gscale = invL * (1.0f / M_DIM);   // (acc/M)/L

  const size_t idx = (size_t)b * D_DIM + d0;
  const float4 zv = *(const float4*)(z + idx);
  const float4 wv = *(const float4*)(w + idx);
  float4 wn, zn;
  {
    float u = zv.x - acc.x * gscale; float a = fabsf(u) - thr;
    wn.x = a > 0.f ? copysignf(a, u) : 0.f;
    u = zv.y - acc.y * gscale; a = fabsf(u) - thr;
    wn.y = a > 0.f ? copysignf(a, u) : 0.f;
    u = zv.z - acc.z * gscale; a = fabsf(u) - thr;
    wn.z = a > 0.f ? copysignf(a, u) : 0.f;
    u = zv.w - acc.w * gscale; a = fabsf(u) - thr;
    wn.w = a > 0.f ? copysignf(a, u) : 0.f;
  }
  zn.x = wn.x + coef * (wn.x - wv.x);
  zn.y = wn.y + coef * (wn.y - wv.y);
  zn.z = wn.z + coef * (wn.z - wv.z);
  zn.w = wn.w + coef * (wn.w - wv.w);
  *(float4*)(w + idx) = wn;
  *(float4*)(z + idx) = zn;
}

// ---------------------------------------------------------------------------
extern "C" void kernel_launch(void* const* d_in, const int* in_sizes, int n_in,
                              void* d_out, int out_size, void* d_ws, size_t ws_size,
                              hipStream_t stream) {
  (void)in_sizes; (void)n_in; (void)out_size; (void)ws_size;
  const float* x = (const float*)d_in[0];        // (8,4,32,256,2) -> (8,65536)
  const float* A = (const float*)d_in[1];        // (8,256,65536)
  float* w = (float*)d_out;                      // final w lives in d_out

  float* ws = (float*)d_ws;                      // ~4.2 MB of scratch
  float* y  = ws;                                // 2048
  float* p  = ws + 2048;                         // 2048
  float* r  = ws + 4096;                         // 2048
  float* L  = ws + 6144;                         // 8 (+pad)
  float* G  = ws + 6400;                         // 524288
  float* z  = ws + 6400 + 524288;                // 524288

  // One pass over A: y and p0 = A*(ones/sqrt(D))
  prep_rows_kernel<<<BATCH * M_DIM, 256, 0, stream>>>(A, x, y, p);
  // One pass over A: Gram matrix via f32 WMMA (replaces 20 streaming passes)
  gram_kernel<<<(BATCH * M_DIM * M_DIM / 256) / 8, 256, 0, stream>>>(A, G);
  // 10 power iterations in M-space + L
  power_kernel<<<BATCH, 256, 0, stream>>>(G, p, L);
  // w = z = 0
  init_wz_kernel<<<BATCH * D_DIM / 256, 256, 0, stream>>>(w, z);

  // FISTA: momentum scalars are data-independent -> host side (f32 to match ref)
  float t = 1.0f;
  for (int k = 0; k < N_ITERS; ++k) {
    const float tn = 0.5f * (1.0f + sqrtf(1.0f + 4.0f * t * t));
    const float coef = (t - 1.0f) / tn;
    t = tn;
    residual_kernel<<<BATCH * M_DIM, 256, 0, stream>>>(A, z, y, r);
    update_kernel<<<BATCH * 64, 256, 0, stream>>>(A, r, L, w, z, coef);
  }
}